// BahdanauAttention_58609123721436
// MI455X (gfx1250) — compile-verified
//
#include <hip/hip_runtime.h>
#include <hip/hip_bf16.h>
#include <math.h>

typedef __attribute__((ext_vector_type(2))) float v2f;
typedef __attribute__((ext_vector_type(8))) float v8f;
typedef int v4i __attribute__((vector_size(16)));   // gcc-style vector to match builtin signature

#define B_ 32
#define L_ 2048
#define E_ 1024
#define U_ 1024
#define KCHUNK 256
#define AS_STRIDE (KCHUNK + 4)   // 260 floats = 1040 B row stride: 16B-aligned, bank-conflict-free
#define RROWS 32                 // L rows per block (2 M-fragments per wave, B frags reused)

#if __has_builtin(__builtin_amdgcn_global_load_async_to_lds_b128) && \
    __has_builtin(__builtin_amdgcn_s_wait_asynccnt)
#define HAVE_ASYNC_LDS 1
typedef v4i __attribute__((address_space(1)))* gbl_v4i_ptr;
typedef v4i __attribute__((address_space(3)))* lds_v4i_ptr;
#else
#define HAVE_ASYNC_LDS 0
#endif

// ---------------------------------------------------------------------------
// Kernel 1: base[b][u] = Ws_b[u] + Wh_b[u] + (use_cov ? Wc_b[u] : 0)
//                        + sum_e dec_hidden[b][e] * Wh_k[e][u]
// ---------------------------------------------------------------------------
__global__ __launch_bounds__(256)
void dec_base_kernel(const float* __restrict__ dec_hidden,
                     const float* __restrict__ Whk,
                     const float* __restrict__ Wsb,
                     const float* __restrict__ Whb,
                     const float* __restrict__ Wcb,
                     const int*   __restrict__ use_cov_p,
                     float*       __restrict__ base)
{
    const int u = blockIdx.x * 256 + threadIdx.x;
    const int b = blockIdx.y;
    float acc = Wsb[u] + Whb[u] + (use_cov_p[0] ? Wcb[u] : 0.0f);
    const float* dh = dec_hidden + (size_t)b * E_;
#pragma unroll 8
    for (int e = 0; e < E_; ++e)
        acc = fmaf(dh[e], Whk[(size_t)e * U_ + u], acc);
    base[(size_t)b * U_ + u] = acc;
}

// ---------------------------------------------------------------------------
// Kernel 2: fused  enc@Ws_k (+bias, +coverage term) -> tanh -> @V_k -> masked
// score.  One block = one (b, 32-row L tile).  8 waves, each owns 8 U-tiles
// and 2 M-fragments; the B fragment is loaded once and fed to 2 WMMAs.
// WMMA: V_WMMA_F32_16X16X4_F32, K looped over E in LDS-staged chunks.
// A-strip staged with GLOBAL_LOAD_ASYNC_TO_LDS_B128 (ASYNCcnt) when available.
// ---------------------------------------------------------------------------
__global__ __launch_bounds__(256)
void score_kernel(const float* __restrict__ enc,
                  const float* __restrict__ Wsk,
                  const float* __restrict__ base,
                  const float* __restrict__ prev_cov,
                  const float* __restrict__ Wck,
                  const float* __restrict__ Vk,
                  const float* __restrict__ Vb,
                  const int*   __restrict__ mask,
                  const int*   __restrict__ use_cov_p,
                  float*       __restrict__ scores)
{
    __shared__ float As[RROWS][AS_STRIDE];
    __shared__ float sScore[RROWS];
    __shared__ float sPC[RROWS];

    const int b    = blockIdx.y;
    const int l0   = blockIdx.x * RROWS;
    const int tid  = threadIdx.x;
    const int wave = tid >> 5;
    const int lane = tid & 31;
    const int half = lane >> 4;        // which half-wave (K offset for A/B frags)
    const int n    = lane & 15;        // N index of this lane in the C tile
    const int use_cov = use_cov_p[0];

    if (tid < RROWS) {
        sScore[tid] = 0.0f;
        sPC[tid]    = prev_cov[(size_t)b * L_ + l0 + tid];
    }

    const float* Arow = enc + ((size_t)b * L_ + l0) * E_;

    v8f acc0[8], acc1[8];
#pragma unroll
    for (int t = 0; t < 8; ++t) {
        v8f z = {0.f,0.f,0.f,0.f,0.f,0.f,0.f,0.f};
        acc0[t] = z;
        acc1[t] = z;
    }

    for (int phase = 0; phase < E_ / KCHUNK; ++phase) {
        const int kbase = phase * KCHUNK;
        __syncthreads();
        // stage RROWS x KCHUNK fp32 A-strip into LDS (16 B per lane per issue)
#if HAVE_ASYNC_LDS
#pragma unroll
        for (int i = tid; i < RROWS * (KCHUNK / 4); i += 256) {
            const int r  = i / (KCHUNK / 4);
            const int c4 = i % (KCHUNK / 4);
            __builtin_amdgcn_global_load_async_to_lds_b128(
                (gbl_v4i_ptr)(Arow + (size_t)r * E_ + kbase + c4 * 4),
                (lds_v4i_ptr)&As[r][c4 * 4],
                0, 0);
        }
        __builtin_amdgcn_s_wait_asynccnt(0);
#else
#pragma unroll
        for (int i = tid; i < RROWS * (KCHUNK / 4); i += 256) {
            const int r  = i / (KCHUNK / 4);
            const int c4 = i % (KCHUNK / 4);
            float4 v = ((const float4*)(Arow + (size_t)r * E_ + kbase))[c4];
            *(float4*)&As[r][c4 * 4] = v;
        }
#endif
        if (phase + 1 < E_ / KCHUNK) {
            // warm the next A chunk (global_prefetch_b8)
            __builtin_prefetch(Arow + (size_t)(tid & 31) * E_ + kbase + KCHUNK +
                               ((tid >> 5) << 5), 0, 3);
        }
        __syncthreads();

#pragma unroll
        for (int t = 0; t < 8; ++t) {
            const int u0 = ((wave << 3) + t) << 4;
            // B fragment rows: half-wave 0 -> K+0/K+1, half-wave 1 -> K+2/K+3
            const float* Bp = Wsk + ((size_t)(kbase + 2 * half)) * U_ + u0 + n;
#pragma unroll 4
            for (int k = 0; k < KCHUNK; k += 4) {
                v2f bb;
                bb.x = Bp[0];
                bb.y = Bp[U_];
                Bp += 4 * (size_t)U_;
                v2f a0, a1;
                a0.x = As[n][k + 2 * half];
                a0.y = As[n][k + 2 * half + 1];
                a1.x = As[n + 16][k + 2 * half];
                a1.y = As[n + 16][k + 2 * half + 1];
                acc0[t] = __builtin_amdgcn_wmma_f32_16x16x4_f32(
                    false, a0, false, bb, (short)0, acc0[t], false, false);
                acc1[t] = __builtin_amdgcn_wmma_f32_16x16x4_f32(
                    false, a1, false, bb, (short)0, acc1[t], false, false);
            }
        }
    }

    // ---- epilogue: pre = acc + base[u] + pc[l]*Wc_k[u]; tanh; dot with V ----
    float partial0[8], partial1[8];
#pragma unroll
    for (int r = 0; r < 8; ++r) { partial0[r] = 0.0f; partial1[r] = 0.0f; }

#pragma unroll
    for (int t = 0; t < 8; ++t) {
        const int u0 = ((wave << 3) + t) << 4;
        const float bsv = base[(size_t)b * U_ + u0 + n];
        const float wcv = use_cov ? Wck[u0 + n] : 0.0f;
        const float vkv = Vk[u0 + n];
#pragma unroll
        for (int r = 0; r < 8; ++r) {
            // C layout: VGPR r holds M=r (lanes 0-15) / M=8+r (lanes 16-31)
            const float pre0 = acc0[t][r] + bsv + wcv * sPC[r + 8 * half];
            partial0[r] += tanhf(pre0) * vkv;
            const float pre1 = acc1[t][r] + bsv + wcv * sPC[16 + r + 8 * half];
            partial1[r] += tanhf(pre1) * vkv;
        }
    }
    // reduce over the 16 N-lanes within each half-wave
#pragma unroll
    for (int r = 0; r < 8; ++r) {
#pragma unroll
        for (int off = 1; off < 16; off <<= 1) {
            partial0[r] += __shfl_xor(partial0[r], off, 32);
            partial1[r] += __shfl_xor(partial1[r], off, 32);
        }
    }
    if (n == 0) {
#pragma unroll
        for (int r = 0; r < 8; ++r) {
            atomicAdd(&sScore[r + 8 * half], partial0[r]);
            atomicAdd(&sScore[16 + r + 8 * half], partial1[r]);
        }
    }
    __syncthreads();
    if (tid < RROWS) {
        const float s = (sScore[tid] + Vb[0]) *
                        (float)mask[(size_t)b * L_ + l0 + tid];
        scores[(size_t)b * L_ + l0 + tid] = s;
    }
}

// ---------------------------------------------------------------------------
// Kernel 3: softmax over L, attention/coverage outputs, context reduction.
// Grid (B, E/256): every block redoes the cheap softmax; block ey==0 writes
// attention + coverage; each block owns 256 contiguous E columns of context.
// ---------------------------------------------------------------------------
__global__ __launch_bounds__(256)
void finish_kernel(const float* __restrict__ scores,
                   const float* __restrict__ enc,
                   const float* __restrict__ prev_cov,
                   const int*   __restrict__ use_cov_p,
                   float*       __restrict__ out_ctx,
                   float*       __restrict__ out_attn,
                   float*       __restrict__ out_cov)
{
    __shared__ float sAttn[L_];
    __shared__ float red[8];
    const int b    = blockIdx.x;
    const int ey   = blockIdx.y;
    const int tid  = threadIdx.x;
    const int lane = tid & 31;
    const int wave = tid >> 5;

    // ---- max ----
    float mx = -INFINITY;
    for (int l = tid; l < L_; l += 256)
        mx = fmaxf(mx, scores[(size_t)b * L_ + l]);
#pragma unroll
    for (int off = 16; off > 0; off >>= 1)
        mx = fmaxf(mx, __shfl_xor(mx, off, 32));
    if (lane == 0) red[wave] = mx;
    __syncthreads();
    mx = red[0];
#pragma unroll
    for (int i = 1; i < 8; ++i) mx = fmaxf(mx, red[i]);
    __syncthreads();

    // ---- exp + sum ----
    float sum = 0.0f;
    for (int l = tid; l < L_; l += 256) {
        const float e = expf(scores[(size_t)b * L_ + l] - mx);
        sAttn[l] = e;
        sum += e;
    }
#pragma unroll
    for (int off = 16; off > 0; off >>= 1)
        sum += __shfl_xor(sum, off, 32);
    if (lane == 0) red[wave] = sum;
    __syncthreads();
    sum = 0.0f;
#pragma unroll
    for (int i = 0; i < 8; ++i) sum += red[i];
    const float inv = 1.0f / sum;

    const int use_cov = use_cov_p[0];
    for (int l = tid; l < L_; l += 256) {
        const float a = sAttn[l] * inv;
        sAttn[l] = a;
        if (ey == 0) {
            out_attn[(size_t)b * L_ + l] = a;
            const float pc = prev_cov[(size_t)b * L_ + l];
            out_cov[(size_t)b * L_ + l] = use_cov ? (a + pc) : a;
        }
    }
    __syncthreads();

    // ---- context: this block's 256 E columns ----
    const int e0 = ey * 256 + tid;
    float acc = 0.0f;
    const float* ep = enc + ((size_t)b * L_) * E_ + e0;
#pragma unroll 4
    for (int l = 0; l < L_; ++l)
        acc = fmaf(sAttn[l], ep[(size_t)l * E_], acc);
    out_ctx[(size_t)b * E_ + e0] = acc;
}

// ---------------------------------------------------------------------------
extern "C" void kernel_launch(void* const* d_in, const int* in_sizes, int n_in,
                              void* d_out, int out_size, void* d_ws, size_t ws_size,
                              hipStream_t stream)
{
    const float* dec_hidden = (const float*)d_in[0];
    const float* enc        = (const float*)d_in[1];
    const float* prev_cov   = (const float*)d_in[2];
    const float* Wsk        = (const float*)d_in[3];
    const float* Wsb        = (const float*)d_in[4];
    const float* Whk        = (const float*)d_in[5];
    const float* Whb        = (const float*)d_in[6];
    const float* Wck        = (const float*)d_in[7];
    const float* Wcb        = (const float*)d_in[8];
    const float* Vk         = (const float*)d_in[9];
    const float* Vb         = (const float*)d_in[10];
    const int*   mask       = (const int*)d_in[11];
    const int*   use_cov    = (const int*)d_in[12];

    float* base   = (float*)d_ws;                  // B*U
    float* scores = base + (size_t)B_ * U_;        // B*L

    float* out_ctx  = (float*)d_out;               // B*E
    float* out_attn = out_ctx + (size_t)B_ * E_;   // B*L
    float* out_cov  = out_attn + (size_t)B_ * L_;  // B*L

    dec_base_kernel<<<dim3(U_ / 256, B_), 256, 0, stream>>>(
        dec_hidden, Whk, Wsb, Whb, Wcb, use_cov, base);
    score_kernel<<<dim3(L_ / RROWS, B_), 256, 0, stream>>>(
        enc, Wsk, base, prev_cov, Wck, Vk, Vb, mask, use_cov, scores);
    finish_kernel<<<dim3(B_, E_ / 256), 256, 0, stream>>>(
        scores, enc, prev_cov, use_cov, out_ctx, out_attn, out_cov);
}